// SVAE_LDS_30021821399700
// MI455X (gfx1250) — compile-verified
//
#include <hip/hip_runtime.h>
#include <cstdint>

#define LATENT 4
#define OBS    10
#define BSZ    256
#define TT     2048
#define NC     256            // time chunks for the parallel scan
#define CL     (TT / NC)      // 8 steps per chunk

// ---------------- workspace layout (float offsets) ----------------
static constexpr size_t WS_ZMU  = 0;                                   // (b,t,4)
static constexpr size_t WS_ZHAT = WS_ZMU  + (size_t)BSZ * TT * 4;      // (b,t,4)
static constexpr size_t WS_PT   = WS_ZHAT + (size_t)BSZ * TT * 4;      // per-t: K16,G16,c4,shift4,pcov16,qcov16 = 72
static constexpr size_t WS_GHAT = WS_PT   + (size_t)TT * 72;           // NC x 16
static constexpr size_t WS_IHAT = WS_GHAT + (size_t)NC * 16;           // NC x BSZ x 4
static constexpr size_t WS_S0   = WS_IHAT + (size_t)NC * BSZ * 4;      // NC x BSZ x 4

// ---------------- output layout (float offsets, tuple order) -------
static constexpr size_t O_XREC = 0;                                    // (bs,T,10)
static constexpr size_t O_QMU  = O_XREC + (size_t)BSZ * TT * OBS;      // (T,bs,4)
static constexpr size_t O_QCOV = O_QMU  + (size_t)TT * BSZ * 4;        // (T,bs,4,4)
static constexpr size_t O_PMU  = O_QCOV + (size_t)TT * BSZ * 16;       // (T,bs,4)
static constexpr size_t O_PCOV = O_PMU  + (size_t)TT * BSZ * 4;        // (T,bs,4,4)

typedef float v2f __attribute__((ext_vector_type(2)));
typedef float v8f __attribute__((ext_vector_type(8)));

// D = A(16x4,f32) x B(4x16,f32) + C(16x16,f32) : V_WMMA_F32_16X16X4_F32
__device__ __forceinline__ v8f wmma4(v2f a, v2f b, v8f c) {
  return __builtin_amdgcn_wmma_f32_16x16x4_f32(false, a, false, b, (short)0, c,
                                               false, false);
}

// Intra-wave LDS ordering: LDS is in-order per wave on CDNA5; the fence +
// wave_barrier stops the compiler from reordering ds ops across the point.
__device__ __forceinline__ void lds_fence() {
  __builtin_amdgcn_fence(__ATOMIC_ACQ_REL, "wavefront");
  __builtin_amdgcn_wave_barrier();
}

// =====================================================================
// Kernel 1: encoder.  z_mu = relu(x W1 + b1) Wx + bx, fused, WMMA f32.
// Tile = 16 consecutive t of one batch row (rows contiguous: 640B of x).
// =====================================================================
__global__ __launch_bounds__(256) void encoder_kernel(
    const float* __restrict__ x, const float* __restrict__ w1,
    const float* __restrict__ b1, const float* __restrict__ wx,
    const float* __restrict__ bx, float* __restrict__ zmu) {
  __shared__ float sh[8][16 * 8];  // per-wave staging
  const int lane  = threadIdx.x & 31;
  const int wv    = threadIdx.x >> 5;
  const int n     = lane & 15;          // C-layout column / B-layout column
  const int m     = lane & 15;          // A-layout row
  const int kbase = (lane >> 4) * 2;    // A/B K pair base
  const bool hi   = lane >= 16;         // upper half-wave (K pairs 2,3)
  float* hsh = sh[wv];

  // B operands for GEMM1: W1 (10x6) zero-padded to 12x16, 3 K-tiles of 4.
  v2f B1[3];
#pragma unroll
  for (int kt = 0; kt < 3; ++kt) {
    int k0 = kt * 4 + kbase;
    B1[kt].x = (k0 < 10 && n < 6) ? w1[k0 * 6 + n] : 0.f;
    B1[kt].y = (k0 + 1 < 10 && n < 6) ? w1[(k0 + 1) * 6 + n] : 0.f;
  }
  // B operands for GEMM2: Wx (6x4) zero-padded to 8x16, 2 K-tiles.
  v2f B2[2];
#pragma unroll
  for (int kt = 0; kt < 2; ++kt) {
    int k0 = kt * 4 + kbase;
    B2[kt].x = (k0 < 6 && n < 4) ? wx[k0 * 4 + n] : 0.f;
    B2[kt].y = (k0 + 1 < 6 && n < 4) ? wx[(k0 + 1) * 4 + n] : 0.f;
  }
  const float bias1 = (n < 6) ? b1[n] : 0.f;
  const float bias2 = (n < 4) ? bx[n] : 0.f;

  const int nTiles = BSZ * (TT / 16);   // 32768
  const int wid    = blockIdx.x * (blockDim.x >> 5) + wv;
  const int nW     = gridDim.x * (blockDim.x >> 5);

  for (int tile = wid; tile < nTiles; tile += nW) {
    const int bi = tile >> 7;           // T/16 = 128 tiles per batch row
    const int tg = tile & 127;
    const float* xrow = x + ((size_t)bi * TT + tg * 16 + m) * OBS;
    {  // prefetch this wave's actual next grid-stride tile (speculative-safe)
      const int nt  = tile + nW;
      const int bi2 = nt >> 7, tg2 = nt & 127;
      __builtin_prefetch(x + ((size_t)bi2 * TT + tg2 * 16 + m) * OBS, 0, 1);
    }

    v8f acc;
#pragma unroll
    for (int r = 0; r < 8; ++r) acc[r] = bias1;
    // K-tiles 0,1: fully in-range vector loads (8B aligned: 40r + {0,8,16,24})
    {
      float2 a01 = *(const float2*)(xrow + kbase);
      v2f a; a.x = a01.x; a.y = a01.y;
      acc = wmma4(a, B1[0], acc);
    }
    {
      float2 a01 = *(const float2*)(xrow + 4 + kbase);
      v2f a; a.x = a01.x; a.y = a01.y;
      acc = wmma4(a, B1[1], acc);
    }
    // K-tile 2: lanes>=16 map to K=10,11 (padding).  The address xrow+8 is
    // valid for every lane, so load unconditionally and select-zero: no
    // exec-mask branches in the hot loop.
    {
      float2 a01 = *(const float2*)(xrow + 8);
      v2f a; a.x = hi ? 0.f : a01.x; a.y = hi ? 0.f : a01.y;
      acc = wmma4(a, B1[2], acc);
    }
    // relu + C-layout -> LDS (16x8 row-major).  Columns 6,7 of acc are
    // exactly zero by construction (zero-padded B1 columns, zero bias), so
    // no column select is needed.
    if (n < 8) {
#pragma unroll
      for (int r = 0; r < 8; ++r)
        hsh[((lane >> 4) * 8 + r) * 8 + n] = fmaxf(acc[r], 0.f);
    }
    lds_fence();
    v8f acc2;
#pragma unroll
    for (int r = 0; r < 8; ++r) acc2[r] = bias2;
#pragma unroll
    for (int kt = 0; kt < 2; ++kt) {
      int k0 = kt * 4 + kbase;
      v2f a;
      a.x = hsh[m * 8 + k0];
      a.y = hsh[m * 8 + k0 + 1];
      acc2 = wmma4(a, B2[kt], acc2);
    }
    lds_fence();
    // z tile (16x4) -> LDS -> coalesced float4 rows
    if (n < 4) {
#pragma unroll
      for (int r = 0; r < 8; ++r) hsh[((lane >> 4) * 8 + r) * 4 + n] = acc2[r];
    }
    lds_fence();
    if (lane < 16) {
      float4 v = *(float4*)&hsh[lane * 4];
      *(float4*)&zmu[((size_t)bi * TT + tg * 16 + lane) * 4] = v;
    }
    lds_fence();
  }
}

// =====================================================================
// Kernel 2: batch-independent Riccati / Cholesky / RNG precompute.
// Single wave; 2048 serial steps of 4x4 algebra (tiny vs memory phase).
// =====================================================================
__device__ inline void inv4(const float* m, float* out) {
  float inv[16];
  inv[0]  =  m[5]*m[10]*m[15] - m[5]*m[11]*m[14] - m[9]*m[6]*m[15] + m[9]*m[7]*m[14] + m[13]*m[6]*m[11] - m[13]*m[7]*m[10];
  inv[4]  = -m[4]*m[10]*m[15] + m[4]*m[11]*m[14] + m[8]*m[6]*m[15] - m[8]*m[7]*m[14] - m[12]*m[6]*m[11] + m[12]*m[7]*m[10];
  inv[8]  =  m[4]*m[9]*m[15]  - m[4]*m[11]*m[13] - m[8]*m[5]*m[15] + m[8]*m[7]*m[13] + m[12]*m[5]*m[11] - m[12]*m[7]*m[9];
  inv[12] = -m[4]*m[9]*m[14]  + m[4]*m[10]*m[13] + m[8]*m[5]*m[14] - m[8]*m[6]*m[13] - m[12]*m[5]*m[10] + m[12]*m[6]*m[9];
  inv[1]  = -m[1]*m[10]*m[15] + m[1]*m[11]*m[14] + m[9]*m[2]*m[15] - m[9]*m[3]*m[14] - m[13]*m[2]*m[11] + m[13]*m[3]*m[10];
  inv[5]  =  m[0]*m[10]*m[15] - m[0]*m[11]*m[14] - m[8]*m[2]*m[15] + m[8]*m[3]*m[14] + m[12]*m[2]*m[11] - m[12]*m[3]*m[10];
  inv[9]  = -m[0]*m[9]*m[15]  + m[0]*m[11]*m[13] + m[8]*m[1]*m[15] - m[8]*m[3]*m[13] - m[12]*m[1]*m[11] + m[12]*m[3]*m[9];
  inv[13] =  m[0]*m[9]*m[14]  - m[0]*m[10]*m[13] - m[8]*m[1]*m[14] + m[8]*m[2]*m[13] + m[12]*m[1]*m[10] - m[12]*m[2]*m[9];
  inv[2]  =  m[1]*m[6]*m[15]  - m[1]*m[7]*m[14]  - m[5]*m[2]*m[15] + m[5]*m[3]*m[14] + m[13]*m[2]*m[7]  - m[13]*m[3]*m[6];
  inv[6]  = -m[0]*m[6]*m[15]  + m[0]*m[7]*m[14]  + m[4]*m[2]*m[15] - m[4]*m[3]*m[14] - m[12]*m[2]*m[7]  + m[12]*m[3]*m[6];
  inv[10] =  m[0]*m[5]*m[15]  - m[0]*m[7]*m[13]  - m[4]*m[1]*m[15] + m[4]*m[3]*m[13] + m[12]*m[1]*m[7]  - m[12]*m[3]*m[5];
  inv[14] = -m[0]*m[5]*m[14]  + m[0]*m[6]*m[13]  + m[4]*m[1]*m[14] - m[4]*m[2]*m[13] - m[12]*m[1]*m[6]  + m[12]*m[2]*m[5];
  inv[3]  = -m[1]*m[6]*m[11]  + m[1]*m[7]*m[10]  + m[5]*m[2]*m[11] - m[5]*m[3]*m[10] - m[9]*m[2]*m[7]   + m[9]*m[3]*m[6];
  inv[7]  =  m[0]*m[6]*m[11]  - m[0]*m[7]*m[10]  - m[4]*m[2]*m[11] + m[4]*m[3]*m[10] + m[8]*m[2]*m[7]   - m[8]*m[3]*m[6];
  inv[11] = -m[0]*m[5]*m[11]  + m[0]*m[7]*m[9]   + m[4]*m[1]*m[11] - m[4]*m[3]*m[9]  - m[8]*m[1]*m[7]   + m[8]*m[3]*m[5];
  inv[15] =  m[0]*m[5]*m[10]  - m[0]*m[6]*m[9]   - m[4]*m[1]*m[10] + m[4]*m[2]*m[9]  + m[8]*m[1]*m[6]   - m[8]*m[2]*m[5];
  float det = m[0]*inv[0] + m[1]*inv[4] + m[2]*inv[8] + m[3]*inv[12];
  float rd = 1.0f / det;
#pragma unroll
  for (int i = 0; i < 16; ++i) out[i] = inv[i] * rd;
}

__device__ inline unsigned pcg_hash(unsigned x) {
  x = x * 747796405u + 2891336453u;
  unsigned w = ((x >> ((x >> 28) + 4u)) ^ x) * 277803737u;
  return (w >> 22) ^ w;
}

__global__ __launch_bounds__(32) void precompute_kernel(
    const float* __restrict__ Ag, const float* __restrict__ bg,
    const float* __restrict__ Qg, const int* __restrict__ seedp,
    float* __restrict__ PT) {
  if (threadIdx.x != 0 || blockIdx.x != 0) return;
  float A[16], Q[16], Qm[16], P[16], bb[4];
#pragma unroll
  for (int i = 0; i < 16; ++i) { A[i] = Ag[i]; Q[i] = Qg[i]; }
#pragma unroll
  for (int i = 0; i < 4; ++i) bb[i] = bg[i];
  for (int i = 0; i < 4; ++i)
    for (int j = 0; j < 4; ++j) {
      float s = 0.f;
      for (int k = 0; k < 4; ++k) s += Q[k * 4 + i] * Q[k * 4 + j];  // Q^T Q
      Qm[i * 4 + j] = s;
      P[i * 4 + j] = (i == j) ? 1.f : 0.f;
    }
  const unsigned sbase = (unsigned)seedp[0] * 0x9E3779B9u + 0x85EBCA6Bu;

  for (int t = 0; t < TT; ++t) {
    float* rec = PT + (size_t)t * 72;
    float AP[16], Pp[16], S[16], Sinv[16], K[16], ImK[16], Pn[16], L[16];
    for (int i = 0; i < 4; ++i)
      for (int j = 0; j < 4; ++j) {
        float s = 0.f;
        for (int k = 0; k < 4; ++k) s += A[i * 4 + k] * P[k * 4 + j];
        AP[i * 4 + j] = s;
      }
    for (int i = 0; i < 4; ++i)
      for (int j = 0; j < 4; ++j) {
        float s = Qm[i * 4 + j];
        for (int k = 0; k < 4; ++k) s += AP[i * 4 + k] * A[j * 4 + k];
        Pp[i * 4 + j] = s;                      // P_pred = A P A^T + Qm
        S[i * 4 + j] = s + ((i == j) ? 1.f : 0.f);
      }
    inv4(S, Sinv);
    for (int i = 0; i < 4; ++i)
      for (int j = 0; j < 4; ++j) {
        float s = 0.f;
        for (int k = 0; k < 4; ++k) s += Pp[i * 4 + k] * Sinv[k * 4 + j];
        K[i * 4 + j] = s;
        ImK[i * 4 + j] = ((i == j) ? 1.f : 0.f) - s;
      }
    for (int i = 0; i < 4; ++i)
      for (int j = 0; j < 4; ++j) {
        float s = 0.f;
        for (int k = 0; k < 4; ++k) s += ImK[i * 4 + k] * Pp[k * 4 + j];
        Pn[i * 4 + j] = s;                      // P = (I-K) P_pred
      }
    // Cholesky of Pn + 1e-3 I (lower)
    for (int i = 0; i < 16; ++i) L[i] = 0.f;
    for (int j = 0; j < 4; ++j) {
      float s = Pn[j * 4 + j] + 1e-3f;
      for (int k = 0; k < j; ++k) s -= L[j * 4 + k] * L[j * 4 + k];
      float d = sqrtf(fmaxf(s, 1e-20f));
      L[j * 4 + j] = d;
      float rdj = 1.f / d;
      for (int i = j + 1; i < 4; ++i) {
        float v = Pn[i * 4 + j];
        for (int k = 0; k < j; ++k) v -= L[i * 4 + k] * L[j * 4 + k];
        L[i * 4 + j] = v * rdj;
      }
    }
    // u ~ N(0,1)^4 (counter-based Box-Muller; shared across batch)
    float u[4];
#pragma unroll
    for (int p = 0; p < 2; ++p) {
      unsigned h1 = pcg_hash(sbase ^ (unsigned)(t * 4 + p * 2));
      unsigned h2 = pcg_hash(sbase ^ (unsigned)(t * 4 + p * 2 + 1));
      float u1 = ((float)h1 + 0.5f) * 2.3283064e-10f;
      float u2 = ((float)h2 + 0.5f) * 2.3283064e-10f;
      float r = sqrtf(-2.f * logf(u1));
      float a = 6.2831853f * u2;
      u[2 * p] = r * cosf(a);
      u[2 * p + 1] = r * sinf(a);
    }
    float shift[4], G[16], cc[4];
    for (int j = 0; j < 4; ++j) {
      float s = 0.f;
      for (int i = 0; i < 4; ++i) s += u[i] * L[i * 4 + j];   // u @ L
      shift[j] = s;
    }
    // mu_t[i] = sum_k mu_{t-1}[k] G[k][i] + (c[i] + sum_j K[i][j] zmu[j])
    for (int k = 0; k < 4; ++k)
      for (int i = 0; i < 4; ++i) {
        float s = 0.f;
        for (int j = 0; j < 4; ++j) s += A[k * 4 + j] * ImK[i * 4 + j];
        G[k * 4 + i] = s;
      }
    for (int i = 0; i < 4; ++i) {
      float s = 0.f;
      for (int j = 0; j < 4; ++j) s += ImK[i * 4 + j] * bb[j];
      cc[i] = s;
    }
#pragma unroll
    for (int i = 0; i < 16; ++i) rec[i] = K[i];
#pragma unroll
    for (int i = 0; i < 16; ++i) rec[16 + i] = G[i];
#pragma unroll
    for (int i = 0; i < 4; ++i)  rec[32 + i] = cc[i];
#pragma unroll
    for (int i = 0; i < 4; ++i)  rec[36 + i] = shift[i];
#pragma unroll
    for (int i = 0; i < 16; ++i) rec[40 + i] = Pp[i];
#pragma unroll
    for (int i = 0; i < 16; ++i) rec[56 + i] = Pn[i];
#pragma unroll
    for (int i = 0; i < 16; ++i) P[i] = Pn[i];
  }
}

// =====================================================================
// Kernel 3: per-chunk partial scan.  Block = chunk c, thread = batch b.
// Computes Ihat_c(b) = chunk run from zero state; thread 0 computes Ghat_c.
// =====================================================================
__global__ __launch_bounds__(256) void scan_chunk_kernel(
    const float* __restrict__ PT, const float* __restrict__ zmu,
    float* __restrict__ Ihat, float* __restrict__ Ghat) {
  __shared__ float sp[CL][36];  // K(16), G(16), c(4) per step
  const int c = blockIdx.x, b = threadIdx.x, t0 = c * CL;
  for (int idx = threadIdx.x; idx < CL * 36; idx += blockDim.x)
    sp[idx / 36][idx % 36] = PT[(size_t)(t0 + idx / 36) * 72 + idx % 36];
  __syncthreads();

  float s[4] = {0.f, 0.f, 0.f, 0.f};
#pragma unroll
  for (int tl = 0; tl < CL; ++tl) {
    const float* K = &sp[tl][0];
    const float* G = &sp[tl][16];
    const float* cc = &sp[tl][32];
    float4 zv = *(const float4*)&zmu[((size_t)b * TT + t0 + tl) * 4];
    float z[4] = {zv.x, zv.y, zv.z, zv.w};
    float sn[4];
#pragma unroll
    for (int i = 0; i < 4; ++i) {
      float v = cc[i];
#pragma unroll
      for (int j = 0; j < 4; ++j) v += K[i * 4 + j] * z[j];
#pragma unroll
      for (int k = 0; k < 4; ++k) v += s[k] * G[k * 4 + i];
      sn[i] = v;
    }
#pragma unroll
    for (int i = 0; i < 4; ++i) s[i] = sn[i];
  }
  float4 o; o.x = s[0]; o.y = s[1]; o.z = s[2]; o.w = s[3];
  *(float4*)&Ihat[((size_t)c * BSZ + b) * 4] = o;

  if (threadIdx.x == 0) {  // chunk matrix product Ghat_c = G_t0 ... G_t0+CL-1
    float Gh[16];
#pragma unroll
    for (int i = 0; i < 16; ++i) Gh[i] = (i % 5 == 0) ? 1.f : 0.f;
    for (int tl = 0; tl < CL; ++tl) {
      const float* G = &sp[tl][16];
      float Nn[16];
      for (int k = 0; k < 4; ++k)
        for (int i = 0; i < 4; ++i) {
          float v = 0.f;
          for (int mq = 0; mq < 4; ++mq) v += Gh[k * 4 + mq] * G[mq * 4 + i];
          Nn[k * 4 + i] = v;
        }
      for (int i = 0; i < 16; ++i) Gh[i] = Nn[i];
    }
    for (int i = 0; i < 16; ++i) Ghat[(size_t)c * 16 + i] = Gh[i];
  }
}

// =====================================================================
// Kernel 4: sequential chunk combine (tiny).  Thread = batch b.
// =====================================================================
__global__ __launch_bounds__(256) void combine_kernel(
    const float* __restrict__ Ghat, const float* __restrict__ Ihat,
    float* __restrict__ S0) {
  const int b = threadIdx.x;
  float s[4] = {0.f, 0.f, 0.f, 0.f};
  for (int c = 0; c < NC; ++c) {
    float4 o; o.x = s[0]; o.y = s[1]; o.z = s[2]; o.w = s[3];
    *(float4*)&S0[((size_t)c * BSZ + b) * 4] = o;
    float4 ih = *(const float4*)&Ihat[((size_t)c * BSZ + b) * 4];
    float ihv[4] = {ih.x, ih.y, ih.z, ih.w};
    float sn[4];
#pragma unroll
    for (int i = 0; i < 4; ++i) {
      float v = ihv[i];
#pragma unroll
      for (int k = 0; k < 4; ++k) v += s[k] * Ghat[(size_t)c * 16 + k * 4 + i];
      sn[i] = v;
    }
#pragma unroll
    for (int i = 0; i < 4; ++i) s[i] = sn[i];
  }
}

// =====================================================================
// Kernel 5: replay.  Block = chunk, thread = batch b.  Produces p_mu,
// q_mu, zhat, and the batch-broadcast p_cov / q_cov streaming writes.
// =====================================================================
__global__ __launch_bounds__(256) void replay_kernel(
    const float* __restrict__ PT, const float* __restrict__ zmu,
    const float* __restrict__ S0, const float* __restrict__ Ag,
    const float* __restrict__ bg, float* __restrict__ zhat,
    float* __restrict__ out) {
  __shared__ float sp[CL][72];
  __shared__ float sA[16];
  __shared__ float sb[4];
  const int c = blockIdx.x, b = threadIdx.x, t0 = c * CL;
  for (int idx = threadIdx.x; idx < CL * 72; idx += blockDim.x)
    sp[idx / 72][idx % 72] = PT[(size_t)(t0 + idx / 72) * 72 + idx % 72];
  if (threadIdx.x < 16) sA[threadIdx.x] = Ag[threadIdx.x];
  if (threadIdx.x < 4) sb[threadIdx.x] = bg[threadIdx.x];
  __syncthreads();

  float4 s4 = *(const float4*)&S0[((size_t)c * BSZ + b) * 4];
  float s[4] = {s4.x, s4.y, s4.z, s4.w};

#pragma unroll
  for (int tl = 0; tl < CL; ++tl) {
    const int t = t0 + tl;
    const float* K = &sp[tl][0];
    const float* shift = &sp[tl][36];
    const float* pc = &sp[tl][40];
    const float* qc = &sp[tl][56];

    float mp[4];
#pragma unroll
    for (int j = 0; j < 4; ++j) {
      float v = sb[j];
#pragma unroll
      for (int k = 0; k < 4; ++k) v += s[k] * sA[k * 4 + j];
      mp[j] = v;                                     // mu_pred
    }
    { float4 o; o.x = mp[0]; o.y = mp[1]; o.z = mp[2]; o.w = mp[3];
      *(float4*)&out[O_PMU + ((size_t)t * BSZ + b) * 4] = o; }

    float4 zv = *(const float4*)&zmu[((size_t)b * TT + t) * 4];
    float d[4] = {zv.x - mp[0], zv.y - mp[1], zv.z - mp[2], zv.w - mp[3]};
    float mu[4];
#pragma unroll
    for (int i = 0; i < 4; ++i) {
      float v = mp[i];
#pragma unroll
      for (int j = 0; j < 4; ++j) v += K[i * 4 + j] * d[j];
      mu[i] = v;
    }
    { float4 o; o.x = mu[0]; o.y = mu[1]; o.z = mu[2]; o.w = mu[3];
      *(float4*)&out[O_QMU + ((size_t)t * BSZ + b) * 4] = o; }
    { float4 o; o.x = mu[0] + shift[0]; o.y = mu[1] + shift[1];
      o.z = mu[2] + shift[2]; o.w = mu[3] + shift[3];
      *(float4*)&zhat[((size_t)b * TT + t) * 4] = o; }

    // broadcast covariance outputs (uniform LDS reads, coalesced stores)
#pragma unroll
    for (int q = 0; q < 4; ++q) {
      float4 o; o.x = pc[q * 4]; o.y = pc[q * 4 + 1];
      o.z = pc[q * 4 + 2]; o.w = pc[q * 4 + 3];
      *(float4*)&out[O_PCOV + ((size_t)t * BSZ + b) * 16 + q * 4] = o;
    }
#pragma unroll
    for (int q = 0; q < 4; ++q) {
      float4 o; o.x = qc[q * 4]; o.y = qc[q * 4 + 1];
      o.z = qc[q * 4 + 2]; o.w = qc[q * 4 + 3];
      *(float4*)&out[O_QCOV + ((size_t)t * BSZ + b) * 16 + q * 4] = o;
    }
#pragma unroll
    for (int i = 0; i < 4; ++i) s[i] = mu[i];
  }
}

// =====================================================================
// Kernel 6: decoder.  x_recon = relu(zhat W1 + b1) W2 + b2, WMMA f32.
// =====================================================================
__global__ __launch_bounds__(256) void decoder_kernel(
    const float* __restrict__ zhat, const float* __restrict__ w1,
    const float* __restrict__ b1, const float* __restrict__ w2,
    const float* __restrict__ b2, float* __restrict__ xrec) {
  __shared__ float sh[8][16 * 10];
  const int lane  = threadIdx.x & 31;
  const int wv    = threadIdx.x >> 5;
  const int n     = lane & 15;
  const int m     = lane & 15;
  const int kbase = (lane >> 4) * 2;
  float* hsh = sh[wv];

  v2f B1;  // dec_w1 (4x6) padded to 4x16 : single K-tile (K=4)
  B1.x = (n < 6) ? w1[kbase * 6 + n] : 0.f;
  B1.y = (n < 6) ? w1[(kbase + 1) * 6 + n] : 0.f;
  v2f B2[2];  // dec_w2 (6x10) padded to 8x16 : 2 K-tiles
#pragma unroll
  for (int kt = 0; kt < 2; ++kt) {
    int k0 = kt * 4 + kbase;
    B2[kt].x = (k0 < 6 && n < 10) ? w2[k0 * 10 + n] : 0.f;
    B2[kt].y = (k0 + 1 < 6 && n < 10) ? w2[(k0 + 1) * 10 + n] : 0.f;
  }
  const float bias1 = (n < 6) ? b1[n] : 0.f;
  const float bias2 = (n < 10) ? b2[n] : 0.f;

  const int nTiles = BSZ * (TT / 16);
  const int wid    = blockIdx.x * (blockDim.x >> 5) + wv;
  const int nW     = gridDim.x * (blockDim.x >> 5);

  for (int tile = wid; tile < nTiles; tile += nW) {
    const int bi = tile >> 7;
    const int tg = tile & 127;
    const float* zrow = zhat + ((size_t)bi * TT + tg * 16 + m) * 4;
    {  // prefetch the wave's actual next tile
      const int nt  = tile + nW;
      const int bi2 = nt >> 7, tg2 = nt & 127;
      __builtin_prefetch(zhat + ((size_t)bi2 * TT + tg2 * 16 + m) * 4, 0, 1);
    }

    v8f acc;
#pragma unroll
    for (int r = 0; r < 8; ++r) acc[r] = bias1;
    {
      float2 a01 = *(const float2*)(zrow + kbase);  // 8B aligned: 16r + {0,8}
      v2f a; a.x = a01.x; a.y = a01.y;
      acc = wmma4(a, B1, acc);
    }
    // relu; columns 6,7 are exactly zero already (zero-padded B1, zero bias)
    if (n < 8) {
#pragma unroll
      for (int r = 0; r < 8; ++r)
        hsh[((lane >> 4) * 8 + r) * 8 + n] = fmaxf(acc[r], 0.f);
    }
    lds_fence();
    v8f acc2;
#pragma unroll
    for (int r = 0; r < 8; ++r) acc2[r] = bias2;
#pragma unroll
    for (int kt = 0; kt < 2; ++kt) {
      int k0 = kt * 4 + kbase;
      v2f ah;
      ah.x = hsh[m * 8 + k0];
      ah.y = hsh[m * 8 + k0 + 1];
      acc2 = wmma4(ah, B2[kt], acc2);
    }
    lds_fence();
    // result tile 16x10 -> LDS -> coalesced row stores
    if (n < 10) {
#pragma unroll
      for (int r = 0; r < 8; ++r) hsh[((lane >> 4) * 8 + r) * 10 + n] = acc2[r];
    }
    lds_fence();
    if (lane < 16) {
      float* orow = xrec + ((size_t)bi * TT + tg * 16 + lane) * OBS;
#pragma unroll
      for (int q = 0; q < 5; ++q)
        *(float2*)&orow[2 * q] = *(float2*)&hsh[lane * 10 + 2 * q];
    }
    lds_fence();
  }
}

// =====================================================================
extern "C" void kernel_launch(void* const* d_in, const int* in_sizes, int n_in,
                              void* d_out, int out_size, void* d_ws,
                              size_t ws_size, hipStream_t stream) {
  (void)in_sizes; (void)n_in; (void)out_size; (void)ws_size;
  const float* x      = (const float*)d_in[0];
  const float* enc_w1 = (const float*)d_in[1];
  const float* enc_b1 = (const float*)d_in[2];
  const float* enc_wx = (const float*)d_in[3];
  const float* enc_bx = (const float*)d_in[4];
  // d_in[5], d_in[6]: enc_wl / enc_bl — sigma head, unused by the filter
  const float* dec_w1 = (const float*)d_in[7];
  const float* dec_b1 = (const float*)d_in[8];
  const float* dec_w2 = (const float*)d_in[9];
  const float* dec_b2 = (const float*)d_in[10];
  const float* A      = (const float*)d_in[11];
  const float* bv     = (const float*)d_in[12];
  const float* Q      = (const float*)d_in[13];
  const int*   seed   = (const int*)d_in[14];
  float* out = (float*)d_out;
  float* ws  = (float*)d_ws;

  precompute_kernel<<<1, 32, 0, stream>>>(A, bv, Q, seed, ws + WS_PT);
  encoder_kernel<<<1024, 256, 0, stream>>>(x, enc_w1, enc_b1, enc_wx, enc_bx,
                                           ws + WS_ZMU);
  scan_chunk_kernel<<<NC, BSZ, 0, stream>>>(ws + WS_PT, ws + WS_ZMU,
                                            ws + WS_IHAT, ws + WS_GHAT);
  combine_kernel<<<1, BSZ, 0, stream>>>(ws + WS_GHAT, ws + WS_IHAT, ws + WS_S0);
  replay_kernel<<<NC, BSZ, 0, stream>>>(ws + WS_PT, ws + WS_ZMU, ws + WS_S0, A,
                                        bv, ws + WS_ZHAT, out);
  decoder_kernel<<<1024, 256, 0, stream>>>(ws + WS_ZHAT, dec_w1, dec_b1, dec_w2,
                                           dec_b2, out + O_XREC);
}